// NPF_54400055771843
// MI455X (gfx1250) — compile-verified
//
#include <hip/hip_runtime.h>
#include <math.h>

// ---------------------------------------------------------------------------
// Reference collapse:
//   cand[p,b,d] = f[b,d] + mask[p,b,d] * noise[p,b,d] * s[b]
//   S1 = sum_p mask*noise ; S2 = sum_p mask*noise^2
//   var[b,d] = s^2 * (S2/P - (S1/P)^2)          (f cancels)
//   w = 1/(var+eps);  Z[b,d] = w*(P*f + s*S1) / colsum_d(w)
//
// Pipeline:
//   (1) per-row std (LDS tree reduce)
//   (2) RNG + per-element stats (Philox4x32-10 + Box-Muller; VALU/trans bound)
//   (3) partial column sums of w: async global->LDS double-buffered tiles
//       (GLOBAL_LOAD_ASYNC_TO_LDS_B128 / s_wait_asynccnt) feeding
//       V_WMMA_F32_16X16X4_F32 with a ones A-matrix; 512 waves
//   (4) finalize: deterministically fold 8 partials per column, Z = num/colsum
// ---------------------------------------------------------------------------

typedef __attribute__((ext_vector_type(2))) float v2f;
typedef __attribute__((ext_vector_type(8))) float v8f;

#define B_ROWS   1024
#define D_COLS   512
#define TWO_D    1024
#define NPERT    100
#define PRATE    0.2f
#define EPS_F    1e-6f
#define RCHUNKS  8                  // row chunks for partial column sums
#define ROWS_PER_CHUNK (B_ROWS / RCHUNKS)   // 128
#define TILE_ROWS 8                 // rows per async LDS tile
#define NTILES   (ROWS_PER_CHUNK / TILE_ROWS)  // 16

// ---------------- Philox4x32-10 counter RNG --------------------------------
__device__ __forceinline__ void philox_round(unsigned& c0, unsigned& c1,
                                             unsigned& c2, unsigned& c3,
                                             unsigned k0, unsigned k1) {
    const unsigned M0 = 0xD2511F53u, M1 = 0xCD9E8D57u;
    unsigned hi0 = __umulhi(M0, c0), lo0 = M0 * c0;
    unsigned hi1 = __umulhi(M1, c2), lo1 = M1 * c2;
    c0 = hi1 ^ c1 ^ k0;
    c1 = lo1;
    c2 = hi0 ^ c3 ^ k1;
    c3 = lo0;
}

__device__ __forceinline__ void philox10(unsigned ctr0, unsigned ctr1,
                                         unsigned ctr2, unsigned ctr3,
                                         unsigned& o0, unsigned& o1,
                                         unsigned& o2, unsigned& o3) {
    unsigned k0 = 42u, k1 = 0xCAFEF00Du;
    unsigned c0 = ctr0, c1 = ctr1, c2 = ctr2, c3 = ctr3;
#pragma unroll
    for (int r = 0; r < 10; ++r) {
        philox_round(c0, c1, c2, c3, k0, k1);
        k0 += 0x9E3779B9u; k1 += 0xBB67AE85u;
    }
    o0 = c0; o1 = c1; o2 = c2; o3 = c3;
}

__device__ __forceinline__ float u24(unsigned x) {          // (0,1]
    return (float)((x >> 8) + 1u) * (1.0f / 16777216.0f);
}
__device__ __forceinline__ float u24z(unsigned x) {         // [0,1)
    return (float)(x >> 8) * (1.0f / 16777216.0f);
}

// ---------------- Kernel 1: per-row std (ddof=1) over D=512 ----------------
__global__ void row_std_kernel(const float* __restrict__ spa,
                               const float* __restrict__ spe,
                               float* __restrict__ std_out /* [2048] */) {
    const int row  = blockIdx.x;               // 0..2047 (spa rows then spe rows)
    const int tid  = threadIdx.x;              // 256 threads = 8 waves (wave32)
    const float* src = (row < B_ROWS) ? spa : spe;
    const int r = row & (B_ROWS - 1);

    __shared__ float s_sum[256];
    __shared__ float s_sq[256];

    float v0 = src[r * D_COLS + tid];
    float v1 = src[r * D_COLS + 256 + tid];
    s_sum[tid] = v0 + v1;
    s_sq[tid]  = v0 * v0 + v1 * v1;
    __syncthreads();

#pragma unroll
    for (int off = 128; off > 0; off >>= 1) {
        if (tid < off) {
            s_sum[tid] += s_sum[tid + off];
            s_sq[tid]  += s_sq[tid + off];
        }
        __syncthreads();
    }
    if (tid == 0) {
        float sum = s_sum[0], sq = s_sq[0];
        float var = (sq - sum * sum * (1.0f / (float)D_COLS)) *
                    (1.0f / (float)(D_COLS - 1));
        std_out[row] = sqrtf(fmaxf(var, 0.0f));
    }
}

// ---------------- Kernel 2: RNG-dominated per-element stats ----------------
__global__ void perturb_stats_kernel(const float* __restrict__ spa,
                                     const float* __restrict__ spe,
                                     const float* __restrict__ stds,
                                     float* __restrict__ w_out,
                                     float* __restrict__ num_out) {
    const int e = blockIdx.x * blockDim.x + threadIdx.x;   // 0 .. 2^20-1
    const int b    = e >> 10;
    const int d    = e & (TWO_D - 1);
    const int feat = d >> 9;                                // 0: spatial, 1: spectral
    const int c    = d & (D_COLS - 1);

    const float f = feat ? spe[b * D_COLS + c] : spa[b * D_COLS + c];
    const float s = stds[feat * B_ROWS + b];

    float S1 = 0.0f, S2 = 0.0f;
#pragma unroll 2
    for (int pp = 0; pp < NPERT / 2; ++pp) {
        unsigned r0, r1, r2, r3;
        philox10((unsigned)e, (unsigned)pp, (unsigned)feat, 0x9E3779B9u,
                 r0, r1, r2, r3);
        // Box-Muller: two normals (fast trans ops co-execute with VALU)
        float u0 = u24(r0);
        float u1 = u24z(r1);
        float rad = sqrtf(-2.0f * __logf(u0));
        float sn, cs;
        __sincosf(6.28318530717958647f * u1, &sn, &cs);
        float n0 = rad * cs;
        float n1 = rad * sn;
        // Independent mask draws
        float m0 = (u24z(r2) < PRATE) ? 1.0f : 0.0f;
        float m1 = (u24z(r3) < PRATE) ? 1.0f : 0.0f;
        S1 += m0 * n0 + m1 * n1;
        S2 += m0 * n0 * n0 + m1 * n1 * n1;
    }

    const float invP = 1.0f / (float)NPERT;
    const float mu   = S1 * invP;
    const float var  = s * s * (S2 * invP - mu * mu);
    const float w    = 1.0f / (var + EPS_F);
    w_out[e]   = w;
    num_out[e] = w * ((float)NPERT * f + s * S1);
}

// ---------------- Kernel 3: partial column sums via async-LDS + WMMA -------
// 512 blocks x 32 threads (1 wave). Block = (col strip of 16, row chunk of 128).
// Each 8x16 f32 tile (512 B) is fetched by one GLOBAL_LOAD_ASYNC_TO_LDS_B128
// per lane (32 lanes x 16 B = whole tile, ASYNCcnt +1 per tile), double-
// buffered against the WMMA consumer. B operand (4x16): VGPR0 = row 2*half,
// VGPR1 = row 2*half+1, lanes 0-15 = N. A = ones(16x4) so D accumulates
// column sums into every row; lane<16 of C[0] holds the 16 sums.
__device__ __forceinline__ void async_tile_load(unsigned long long wbase,
                                                unsigned row0, unsigned col0,
                                                unsigned lane,
                                                unsigned lds_byte_addr) {
    unsigned row   = lane >> 2;       // 0..7
    unsigned chunk = lane & 3;        // 0..3, 16 B each
    unsigned goff  = ((row0 + row) * (unsigned)TWO_D + col0 + chunk * 4u) * 4u;
    unsigned lds   = lds_byte_addr + (row * 16u + chunk * 4u) * 4u;
    asm volatile("global_load_async_to_lds_b128 %0, %1, %2"
                 :
                 : "v"(lds), "v"(goff), "s"(wbase)
                 : "memory");
}

__global__ void colsum_wmma_kernel(const float* __restrict__ w,
                                   float* __restrict__ partial /* [8][1024] */) {
    __shared__ __align__(16) float tile[2][TILE_ROWS * 16];

    const int strip  = blockIdx.x & 63;         // 16-column strip
    const int rchunk = blockIdx.x >> 6;         // 0..7
    const int col0   = strip * 16;
    const int r0     = rchunk * ROWS_PER_CHUNK;
    const int lane   = threadIdx.x;             // wave32, EXEC all ones
    const int half   = lane >> 4;
    const int n      = lane & 15;

    const unsigned long long wbase = (unsigned long long)(uintptr_t)w;
    const unsigned lbase0 = (unsigned)(uintptr_t)(&tile[0][0]);
    const unsigned lbase1 = (unsigned)(uintptr_t)(&tile[1][0]);

    v2f a; a.x = 1.0f; a.y = 1.0f;              // ones A-matrix
    v8f c = {};

    async_tile_load(wbase, r0, col0, lane, lbase0);
    for (int t = 0; t < NTILES; ++t) {
        if (t + 1 < NTILES) {
            async_tile_load(wbase, r0 + (t + 1) * TILE_ROWS, col0, lane,
                            (t & 1) ? lbase0 : lbase1);
            asm volatile("s_wait_asynccnt 0x1" ::: "memory");
        } else {
            asm volatile("s_wait_asynccnt 0x0" ::: "memory");
        }
        const float* tb = tile[t & 1];
#pragma unroll
        for (int kc = 0; kc < 2; ++kc) {        // two K=4 chunks per 8-row tile
            v2f bm;
            bm.x = tb[(kc * 4 + 2 * half + 0) * 16 + n];
            bm.y = tb[(kc * 4 + 2 * half + 1) * 16 + n];
            c = __builtin_amdgcn_wmma_f32_16x16x4_f32(
                    false, a, false, bm, (short)0, c, false, false);
        }
    }
    if (lane < 16) partial[rchunk * TWO_D + col0 + n] = c[0];
}

// ---------------- Kernel 4: fold partials, Z = num / colsum ----------------
__global__ void finalize_kernel(const float* __restrict__ num,
                                const float* __restrict__ partial,
                                float* __restrict__ z) {
    const int e = blockIdx.x * blockDim.x + threadIdx.x;
    if (e < B_ROWS * TWO_D) {
        __builtin_prefetch(num + e + 8192, 0, 1);
        const int d = e & (TWO_D - 1);
        float cs = 0.0f;
#pragma unroll
        for (int k = 0; k < RCHUNKS; ++k) cs += partial[k * TWO_D + d];
        z[e] = num[e] / cs;
    }
}

// ---------------------------------------------------------------------------
extern "C" void kernel_launch(void* const* d_in, const int* in_sizes, int n_in,
                              void* d_out, int out_size, void* d_ws, size_t ws_size,
                              hipStream_t stream) {
    (void)in_sizes; (void)n_in; (void)out_size; (void)ws_size;
    const float* spa = (const float*)d_in[0];   // [1024, 512] f32
    const float* spe = (const float*)d_in[1];   // [1024, 512] f32
    float* z = (float*)d_out;                   // [1024, 1024] f32

    // Workspace (floats): stds[2048] | w[2^20] | num[2^20] | partial[8*1024]
    float* ws      = (float*)d_ws;
    float* stds    = ws;
    float* wbuf    = ws + 2048;
    float* numbuf  = wbuf + (B_ROWS * TWO_D);
    float* partial = numbuf + (B_ROWS * TWO_D);

    row_std_kernel<<<2 * B_ROWS, 256, 0, stream>>>(spa, spe, stds);
    perturb_stats_kernel<<<(B_ROWS * TWO_D) / 256, 256, 0, stream>>>(
        spa, spe, stds, wbuf, numbuf);
    colsum_wmma_kernel<<<64 * RCHUNKS, 32, 0, stream>>>(wbuf, partial);
    finalize_kernel<<<(B_ROWS * TWO_D) / 256, 256, 0, stream>>>(numbuf, partial, z);
}